// GATReg_42597485641875
// MI455X (gfx1250) — compile-verified
//
#include <hip/hip_runtime.h>
#include <math.h>

#define GAT_N 50000
#define GAT_E 800000
#define NEG_SLOPE 0.2f
#define NEG_BIG  (-3.402823466e38f)

typedef __attribute__((ext_vector_type(2))) float v2f;
typedef __attribute__((ext_vector_type(8))) float v8f;

// ---------------- float atomic max via integer ordering trick ----------------
__device__ __forceinline__ void atomicMaxF(float* addr, float val) {
  if (val >= 0.0f)
    atomicMax((int*)addr, __float_as_int(val));
  else
    atomicMin((unsigned int*)addr, __float_as_uint(val));
}

// ---------------- WMMA fp32 GEMM: C[M,N] = A[M,K] @ B[K,N], row-major -------
// One wave computes a 16x64 strip (4 x 16x16 tiles sharing the A fragment).
// Requires M%16==0, N%64==0, K%4==0.
__global__ void gat_gemm_wmma(const float* __restrict__ A,
                              const float* __restrict__ B,
                              float* __restrict__ C, int M, int N, int K) {
  const int lane = threadIdx.x & 31;
  const int wave = (int)((blockIdx.x * blockDim.x + threadIdx.x) >> 5);
  const int nb = N >> 6;                       // # of 64-wide column blocks
  const int total = (M >> 4) * nb;
  if (wave >= total) return;                   // wave-uniform: EXEC stays all-1s
  const int mt   = wave / nb;
  const int nblk = (wave % nb) << 6;
  const int l15  = lane & 15;
  const int kh   = (lane >> 4) << 1;           // K-half: 0 (lanes 0-15) or 2 (16-31)

  const float* arow  = A + (size_t)(mt * 16 + l15) * K + kh;
  const float* bbase = B + (size_t)kh * N + nblk + l15;

  v8f c0 = {}, c1 = {}, c2 = {}, c3 = {};
  for (int kk = 0; kk < K; kk += 4) {
    v2f a;                                     // A 16x4: v0=K+0, v1=K+1 (per K-half)
    a.x = arow[kk];
    a.y = arow[kk + 1];
    const float* bp = bbase + (size_t)kk * N;
    v2f b0, b1, b2, b3;                        // B 4x16: v0=row kh, v1=row kh+1
    b0.x = bp[0];      b0.y = bp[N + 0];
    b1.x = bp[16];     b1.y = bp[N + 16];
    b2.x = bp[32];     b2.y = bp[N + 32];
    b3.x = bp[48];     b3.y = bp[N + 48];
    c0 = __builtin_amdgcn_wmma_f32_16x16x4_f32(false, a, false, b0, (short)0, c0, false, false);
    c1 = __builtin_amdgcn_wmma_f32_16x16x4_f32(false, a, false, b1, (short)0, c1, false, false);
    c2 = __builtin_amdgcn_wmma_f32_16x16x4_f32(false, a, false, b2, (short)0, c2, false, false);
    c3 = __builtin_amdgcn_wmma_f32_16x16x4_f32(false, a, false, b3, (short)0, c3, false, false);
  }
  // C/D layout: VGPR g -> row mt*16 + g (+8 for lanes 16-31), col = lane&15
  const int rbase = mt * 16 + ((lane >> 4) << 3);
  float* crow = C + (size_t)rbase * N + nblk + l15;
#pragma unroll
  for (int g = 0; g < 8; ++g) {
    crow[(size_t)g * N + 0]  = c0[g];
    crow[(size_t)g * N + 16] = c1[g];
    crow[(size_t)g * N + 32] = c2[g];
    crow[(size_t)g * N + 48] = c3[g];
  }
}

// ---------------- init: zero accumulators, -inf maxima ----------------------
__global__ void gat_init(float* out1, float* den1, float* amax1,
                         float* den2, float* amax2) {
  int i = blockIdx.x * blockDim.x + threadIdx.x;
  if (i < GAT_N * 256) out1[i] = 0.0f;
  if (i < GAT_N * 4) { den1[i] = 0.0f; amax1[i] = NEG_BIG; }
  if (i < GAT_N)     { den2[i] = 0.0f; amax2[i] = NEG_BIG; }
}

__global__ void gat_zero(float* p, int n) {
  int i = blockIdx.x * blockDim.x + threadIdx.x;
  if (i < n) p[i] = 0.0f;
}

// ---------------- fold att_edge through We: ve1[4][16], ve2[16] -------------
__global__ void gat_prevec(const float* __restrict__ We1, const float* __restrict__ attE1,
                           const float* __restrict__ We2, const float* __restrict__ attE2,
                           float* ve1, float* ve2) {
  int t = threadIdx.x;
  if (t < 64) {
    int h = t >> 4, k = t & 15;
    float s = 0.0f;
    for (int c = 0; c < 64; ++c) s += We1[(size_t)k * 256 + h * 64 + c] * attE1[h * 64 + c];
    ve1[h * 16 + k] = s;
  } else if (t < 80) {
    int k = t - 64;
    float s = 0.0f;
    for (int c = 0; c < 64; ++c) s += We2[(size_t)k * 64 + c] * attE2[c];
    ve2[k] = s;
  }
}

// ---------------- per-node logit pieces (layer 1: wave/node, 256 cols) ------
__global__ void gat_node_scores1(const float* __restrict__ h1,
                                 const float* __restrict__ as1, const float* __restrict__ ad1,
                                 float* __restrict__ asrc, float* __restrict__ adst) {
  int wid  = (int)((blockIdx.x * blockDim.x + threadIdx.x) >> 5);
  int lane = threadIdx.x & 31;
  if (wid >= GAT_N) return;
  const float* row = h1 + (size_t)wid * 256;
  int base = lane * 8;                       // 8 cols per lane, head = lane>>3
  float ss = 0.0f, sd = 0.0f;
#pragma unroll
  for (int j = 0; j < 8; ++j) {
    float v = row[base + j];
    ss += v * as1[base + j];
    sd += v * ad1[base + j];
  }
  for (int off = 4; off; off >>= 1) {
    ss += __shfl_down(ss, off, 8);
    sd += __shfl_down(sd, off, 8);
  }
  if ((lane & 7) == 0) {
    int h = lane >> 3;
    asrc[(size_t)wid * 4 + h] = ss;
    adst[(size_t)wid * 4 + h] = sd;
  }
}

// ---------------- per-node logit pieces (layer 2: wave/node, 64 cols) -------
__global__ void gat_node_scores2(const float* __restrict__ h2,
                                 const float* __restrict__ as2, const float* __restrict__ ad2,
                                 float* __restrict__ asrc, float* __restrict__ adst) {
  int wid  = (int)((blockIdx.x * blockDim.x + threadIdx.x) >> 5);
  int lane = threadIdx.x & 31;
  if (wid >= GAT_N) return;
  const float* row = h2 + (size_t)wid * 64;
  int c = lane * 2;
  float ss = row[c] * as2[c] + row[c + 1] * as2[c + 1];
  float sd = row[c] * ad2[c] + row[c + 1] * ad2[c + 1];
  for (int off = 16; off; off >>= 1) {
    ss += __shfl_down(ss, off, 32);
    sd += __shfl_down(sd, off, 32);
  }
  if (lane == 0) { asrc[wid] = ss; adst[wid] = sd; }
}

// ---------------- edge logit pieces for both layers (one edge_attr pass) ----
__global__ void gat_edge_scores(const float* __restrict__ eattr,
                                const float* __restrict__ ve1, const float* __restrict__ ve2,
                                float* __restrict__ ae1, float* __restrict__ ae2) {
  int e = blockIdx.x * blockDim.x + threadIdx.x;
  if (e >= GAT_E) return;
  float xv[16];
#pragma unroll
  for (int k = 0; k < 16; ++k) xv[k] = eattr[(size_t)e * 16 + k];
#pragma unroll
  for (int h = 0; h < 4; ++h) {
    float s = 0.0f;
#pragma unroll
    for (int k = 0; k < 16; ++k) s += xv[k] * ve1[h * 16 + k];
    ae1[(size_t)e * 4 + h] = s;
  }
  float s2 = 0.0f;
#pragma unroll
  for (int k = 0; k < 16; ++k) s2 += xv[k] * ve2[k];
  ae2[e] = s2;
}

// ---------------- alpha = leaky_relu(sum), segment max (layer 1) ------------
__global__ void gat_alpha_max1(const float* __restrict__ asrc, const float* __restrict__ adst,
                               float* __restrict__ ea1, const int* __restrict__ ei,
                               float* amax1) {
  int e = blockIdx.x * blockDim.x + threadIdx.x;
  if (e >= GAT_E) return;
  int s = ei[e], d = ei[GAT_E + e];
#pragma unroll
  for (int h = 0; h < 4; ++h) {
    float a = asrc[(size_t)s * 4 + h] + adst[(size_t)d * 4 + h] + ea1[(size_t)e * 4 + h];
    a = a > 0.0f ? a : NEG_SLOPE * a;
    ea1[(size_t)e * 4 + h] = a;
    atomicMaxF(&amax1[(size_t)d * 4 + h], a);
  }
}

__global__ void gat_exp_sum1(float* __restrict__ ea1, const int* __restrict__ ei,
                             const float* __restrict__ amax1, float* den1) {
  int e = blockIdx.x * blockDim.x + threadIdx.x;
  if (e >= GAT_E) return;
  int d = ei[GAT_E + e];
#pragma unroll
  for (int h = 0; h < 4; ++h) {
    float ex = expf(ea1[(size_t)e * 4 + h] - amax1[(size_t)d * 4 + h]);
    ea1[(size_t)e * 4 + h] = ex;
    atomicAdd(&den1[(size_t)d * 4 + h], ex);
  }
}

// ---------------- message scatter (layer 1): block = one edge, 256 cols -----
__global__ void gat_scatter1(const float* __restrict__ h1, const float* __restrict__ ea1,
                             const float* __restrict__ den1, const int* __restrict__ ei,
                             float* out1) {
  int e = blockIdx.x;
  int t = threadIdx.x;
  int s = ei[e], d = ei[GAT_E + e];
  int h = t >> 6;
  float coeff = ea1[(size_t)e * 4 + h] / (den1[(size_t)d * 4 + h] + 1e-16f);
  atomicAdd(&out1[(size_t)d * 256 + t], h1[(size_t)s * 256 + t] * coeff);
}

__global__ void gat_finish1(float* out1, const float* __restrict__ b1) {
  int i = blockIdx.x * blockDim.x + threadIdx.x;
  if (i >= GAT_N * 256) return;
  float v = out1[i] + b1[i & 255];
  out1[i] = v > 0.0f ? v : 0.0f;
}

// ---------------- layer-2 alpha / softmax / scatter -------------------------
__global__ void gat_alpha_max2(const float* __restrict__ asrc, const float* __restrict__ adst,
                               float* __restrict__ ea2, const int* __restrict__ ei,
                               float* amax2) {
  int e = blockIdx.x * blockDim.x + threadIdx.x;
  if (e >= GAT_E) return;
  int s = ei[e], d = ei[GAT_E + e];
  float a = asrc[s] + adst[d] + ea2[e];
  a = a > 0.0f ? a : NEG_SLOPE * a;
  ea2[e] = a;
  atomicMaxF(&amax2[d], a);
}

__global__ void gat_exp_sum2(float* __restrict__ ea2, const int* __restrict__ ei,
                             const float* __restrict__ amax2, float* den2) {
  int e = blockIdx.x * blockDim.x + threadIdx.x;
  if (e >= GAT_E) return;
  int d = ei[GAT_E + e];
  float ex = expf(ea2[e] - amax2[d]);
  ea2[e] = ex;
  atomicAdd(&den2[d], ex);
}

__global__ void gat_scatter2(const float* __restrict__ h2, const float* __restrict__ ea2,
                             const float* __restrict__ den2, const int* __restrict__ ei,
                             float* agg2) {
  int t = threadIdx.x;
  int e = blockIdx.x * 4 + (t >> 6);
  if (e >= GAT_E) return;
  int c = t & 63;
  int s = ei[e], d = ei[GAT_E + e];
  float coeff = ea2[e] / (den2[d] + 1e-16f);
  atomicAdd(&agg2[(size_t)d * 64 + c], h2[(size_t)s * 64 + c] * coeff);
}

// ---------------- final: relu(agg2 + b2) @ Wlin + blin (wave per node) ------
__global__ void gat_final(const float* __restrict__ agg2, const float* __restrict__ b2,
                          const float* __restrict__ Wlin, const float* __restrict__ blin,
                          float* __restrict__ out) {
  int wid  = (int)((blockIdx.x * blockDim.x + threadIdx.x) >> 5);
  int lane = threadIdx.x & 31;
  if (wid >= GAT_N) return;
  int c = lane * 2;
  float v0 = agg2[(size_t)wid * 64 + c]     + b2[c];
  float v1 = agg2[(size_t)wid * 64 + c + 1] + b2[c + 1];
  v0 = v0 > 0.0f ? v0 : 0.0f;
  v1 = v1 > 0.0f ? v1 : 0.0f;
  float s = v0 * Wlin[c] + v1 * Wlin[c + 1];
  for (int off = 16; off; off >>= 1) s += __shfl_down(s, off, 32);
  if (lane == 0) out[wid] = s + blin[0];
}

// ============================================================================
extern "C" void kernel_launch(void* const* d_in, const int* in_sizes, int n_in,
                              void* d_out, int out_size, void* d_ws, size_t ws_size,
                              hipStream_t stream) {
  const float* x     = (const float*)d_in[0];
  const int*   ei    = (const int*)d_in[1];
  const float* eattr = (const float*)d_in[2];
  const float* W1    = (const float*)d_in[3];
  const float* We1   = (const float*)d_in[4];
  const float* as1   = (const float*)d_in[5];
  const float* ad1   = (const float*)d_in[6];
  const float* attE1 = (const float*)d_in[7];
  const float* b1    = (const float*)d_in[8];
  const float* W2    = (const float*)d_in[9];
  const float* We2   = (const float*)d_in[10];
  const float* as2   = (const float*)d_in[11];
  const float* ad2   = (const float*)d_in[12];
  const float* attE2 = (const float*)d_in[13];
  const float* b2    = (const float*)d_in[14];
  const float* Wlin  = (const float*)d_in[15];
  const float* blin  = (const float*)d_in[16];
  float* out = (float*)d_out;

  // workspace carve-up (floats)
  float* ws = (float*)d_ws;
  size_t o = 0;
  float* h1    = ws + o; o += (size_t)GAT_N * 256;   // layer-1 features (alias: h2)
  float* out1  = ws + o; o += (size_t)GAT_N * 256;   // layer-1 agg   (alias: agg2)
  float* ea1   = ws + o; o += (size_t)GAT_E * 4;     // a_e1 -> alpha1 -> ex1 (in place)
  float* ea2   = ws + o; o += (size_t)GAT_E;         // a_e2 -> alpha2 -> ex2
  float* asrc1 = ws + o; o += (size_t)GAT_N * 4;
  float* adst1 = ws + o; o += (size_t)GAT_N * 4;
  float* amax1 = ws + o; o += (size_t)GAT_N * 4;
  float* den1  = ws + o; o += (size_t)GAT_N * 4;
  float* asrc2 = ws + o; o += (size_t)GAT_N;
  float* adst2 = ws + o; o += (size_t)GAT_N;
  float* amax2 = ws + o; o += (size_t)GAT_N;
  float* den2  = ws + o; o += (size_t)GAT_N;
  float* ve1   = ws + o; o += 64;
  float* ve2   = ws + o; o += 16;
  float* h2   = h1;    // h1 dead after scatter1
  float* agg2 = out1;  // out1 dead after gemm2

  const int TPB = 256;
  // ---- init + tiny precompute ----
  gat_init<<<(GAT_N * 256 + TPB - 1) / TPB, TPB, 0, stream>>>(out1, den1, amax1, den2, amax2);
  gat_prevec<<<1, 128, 0, stream>>>(We1, attE1, We2, attE2, ve1, ve2);
  // ---- layer 1 ----
  {
    int waves = (GAT_N / 16) * (256 / 64);           // 12500
    gat_gemm_wmma<<<(waves * 32 + TPB - 1) / TPB, TPB, 0, stream>>>(x, W1, h1, GAT_N, 256, 128);
  }
  gat_node_scores1<<<GAT_N / 8, TPB, 0, stream>>>(h1, as1, ad1, asrc1, adst1);
  gat_edge_scores<<<(GAT_E + TPB - 1) / TPB, TPB, 0, stream>>>(eattr, ve1, ve2, ea1, ea2);
  gat_alpha_max1<<<(GAT_E + TPB - 1) / TPB, TPB, 0, stream>>>(asrc1, adst1, ea1, ei, amax1);
  gat_exp_sum1<<<(GAT_E + TPB - 1) / TPB, TPB, 0, stream>>>(ea1, ei, amax1, den1);
  gat_scatter1<<<GAT_E, TPB, 0, stream>>>(h1, ea1, den1, ei, out1);
  gat_finish1<<<(GAT_N * 256 + TPB - 1) / TPB, TPB, 0, stream>>>(out1, b1);
  // ---- layer 2 ----
  {
    int waves = (GAT_N / 16) * (64 / 64);            // 3125
    gat_gemm_wmma<<<(waves * 32 + TPB - 1) / TPB, TPB, 0, stream>>>(out1, W2, h2, GAT_N, 64, 256);
  }
  gat_zero<<<(GAT_N * 64 + TPB - 1) / TPB, TPB, 0, stream>>>(agg2, GAT_N * 64);
  gat_node_scores2<<<GAT_N / 8, TPB, 0, stream>>>(h2, as2, ad2, asrc2, adst2);
  gat_alpha_max2<<<(GAT_E + TPB - 1) / TPB, TPB, 0, stream>>>(asrc2, adst2, ea2, ei, amax2);
  gat_exp_sum2<<<(GAT_E + TPB - 1) / TPB, TPB, 0, stream>>>(ea2, ei, amax2, den2);
  gat_scatter2<<<(GAT_E + 3) / 4, TPB, 0, stream>>>(h2, ea2, den2, ei, agg2);
  // ---- output projection ----
  gat_final<<<GAT_N / 8, TPB, 0, stream>>>(agg2, b2, Wlin, blin, out);
}